// EuclideanExperts_66314295050614
// MI455X (gfx1250) — compile-verified
//
#include <hip/hip_runtime.h>
#include <hip/hip_bf16.h>

#define N_NODES 100000
#define N_EDGES 1600000
#define D 128
#define NUM_EXPERTS 8
#define KCHUNK 32
#define N_TILES (N_NODES / 16)          // 6250, exact (N divisible by 16)
#define N_TASKS (N_TILES / 2)           // 3125 double-row tasks (32 rows each)
#define NB ((N_NODES + 255) / 256)      // 391 scan blocks

typedef float v2f __attribute__((ext_vector_type(2)));
typedef float v8f __attribute__((ext_vector_type(8)));

// =================== CSR build: dst-sorted adjacency (built once) ===================

__global__ void count_kernel(const int* __restrict__ dst, int* __restrict__ cnt, int E) {
    int i = blockIdx.x * blockDim.x + threadIdx.x;
    if (i < E) atomicAdd(&cnt[dst[i]], 1);
}

__global__ void scan1_kernel(const int* __restrict__ cnt, int* __restrict__ bsum) {
    __shared__ int s[256];
    int t = threadIdx.x, i = blockIdx.x * 256 + t;
    s[t] = (i < N_NODES) ? cnt[i] : 0;
    __syncthreads();
    for (int off = 128; off > 0; off >>= 1) {
        if (t < off) s[t] += s[t + off];
        __syncthreads();
    }
    if (t == 0) bsum[blockIdx.x] = s[0];
}

__global__ void scan2_kernel(const int* __restrict__ bsum, int* __restrict__ boff,
                             int* __restrict__ rp_total) {
    if (threadIdx.x == 0 && blockIdx.x == 0) {
        int acc = 0;
        for (int j = 0; j < NB; ++j) { boff[j] = acc; acc += bsum[j]; }
        rp_total[0] = acc;  // rp[N_NODES]
    }
}

__global__ void scan3_kernel(const int* __restrict__ cnt, const int* __restrict__ boff,
                             int* __restrict__ rp) {
    __shared__ int s[256];
    int t = threadIdx.x, i = blockIdx.x * 256 + t;
    int val = (i < N_NODES) ? cnt[i] : 0;
    s[t] = val;
    __syncthreads();
    for (int off = 1; off < 256; off <<= 1) {
        int x = (t >= off) ? s[t - off] : 0;
        __syncthreads();
        s[t] += x;
        __syncthreads();
    }
    if (i < N_NODES) rp[i] = boff[blockIdx.x] + s[t] - val;  // exclusive
}

__global__ void fill_kernel(const int* __restrict__ src, const int* __restrict__ dst,
                            const int* __restrict__ rp, int* __restrict__ cursor,
                            int* __restrict__ list, int E) {
    int i = blockIdx.x * blockDim.x + threadIdx.x;
    if (i >= E) return;
    int d = dst[i];
    int pos = atomicAdd(&cursor[d], 1);
    list[rp[d] + pos] = src[i];
}

// =================== mean aggregation via CSR gather (no atomics) ===================

__global__ void gather_mean(const float* __restrict__ feat, const int* __restrict__ list,
                            const int* __restrict__ rp, float* __restrict__ out) {
    int gid = blockIdx.x * 256 + threadIdx.x;
    int n = gid >> 5;
    if (n >= N_NODES) return;
    int lane = gid & 31;
    int beg = rp[n], end = rp[n + 1];
    const int f = lane << 2;
    float4 acc = make_float4(0.f, 0.f, 0.f, 0.f);
    for (int base = beg; base < end; base += 32) {
        int myIdx = (base + lane < end) ? list[base + lane] : 0;
        int cn = end - base; if (cn > 32) cn = 32;
        for (int j = 0; j < cn; ++j) {
            int s = __shfl(myIdx, j, 32);
            const float4 v = *(const float4*)(feat + (size_t)s * D + f);
            acc.x += v.x; acc.y += v.y; acc.z += v.z; acc.w += v.w;
        }
    }
    float inv = 1.0f / fmaxf((float)(end - beg), 1.0f);
    float4 r = make_float4(acc.x * inv, acc.y * inv, acc.z * inv, acc.w * inv);
    *(float4*)(out + (size_t)n * D + f) = r;
}

// =================== fused dual-GEMM: out = Xs@Ws + Xn@Wn + bias ===================
// Block = 8 waves = 4 double-row tasks x 2 column halves (128 rows x 128 cols).
// Each wave: 2 row tiles x 4 col tiles -> 64 accumulator VGPRs (spill-free) while
// every B fragment (one half of a ds_load_2addr_b64) feeds TWO WMMAs.
// Weights staged in LDS K-pair-interleaved with padded stride (288 == 32 mod 64 banks)
// so the two lane halves hit complementary banks: conflict-free b64 reads.

constexpr int SP = 2 * D + 32;  // pair-row stride in floats

template <int MODE>
__global__ void __launch_bounds__(256)
sage_gemm(const float* __restrict__ Xs, const float* __restrict__ Xn,
          const float* __restrict__ Ws, const float* __restrict__ Wn,
          const float* __restrict__ bias, float* __restrict__ outp,
          float* __restrict__ stats) {
    __shared__ float sWs[(KCHUNK / 2) * SP];
    __shared__ float sWn[(KCHUNK / 2) * SP];
    const int tid  = threadIdx.x;
    const int wave = tid >> 5;
    const int lane = tid & 31;
    const int task = blockIdx.x * 4 + (wave >> 1);   // 0 .. 3127
    const bool valid = (task < N_TASKS);             // 3125 valid
    const int row0 = (valid ? task : 0) * 32;        // clamp keeps EXEC all-ones thru WMMA
    const int cb   = (wave & 1) * 64;                // column half base
    const int mcol = lane & 15;                      // A-row / B,C,D-col within tile
    const int koff = (lane >> 4) << 1;               // 0 (lanes 0-15) or 2 (lanes 16-31)

    v8f acc[2][4];
#pragma unroll
    for (int ct = 0; ct < 4; ++ct) {
        float bv = bias[cb + ct * 16 + mcol];
#pragma unroll
        for (int v = 0; v < 8; ++v) { acc[0][ct][v] = bv; acc[1][ct][v] = bv; }
    }

    const float* aS0 = Xs + (size_t)(row0 + mcol) * D + koff;
    const float* aS1 = Xs + (size_t)(row0 + 16 + mcol) * D + koff;
    const float* aN0 = Xn + (size_t)(row0 + mcol) * D + koff;
    const float* aN1 = Xn + (size_t)(row0 + 16 + mcol) * D + koff;

    for (int kb = 0; kb < D; kb += KCHUNK) {
        // stage K-chunk, pair-interleaved: sW[kp*SP + 2c] = {W[2kp][c], W[2kp+1][c]}
        for (int idx = tid; idx < (KCHUNK / 2) * (D / 2); idx += 256) {
            int kp = idx >> 6, c = (idx & 63) << 1;
            const float* g = Ws + (size_t)(kb + 2 * kp) * D + c;
            v2f r0 = *(const v2f*)g;
            v2f r1 = *(const v2f*)(g + D);
            float* s = sWs + kp * SP + 2 * c;
            s[0] = r0[0]; s[1] = r1[0]; s[2] = r0[1]; s[3] = r1[1];
            g = Wn + (size_t)(kb + 2 * kp) * D + c;
            r0 = *(const v2f*)g;
            r1 = *(const v2f*)(g + D);
            s = sWn + kp * SP + 2 * c;
            s[0] = r0[0]; s[1] = r1[0]; s[2] = r0[1]; s[3] = r1[1];
        }
        __syncthreads();
#pragma unroll
        for (int k = 0; k < KCHUNK; k += 4) {
            v2f ax0 = *(const v2f*)(aS0 + kb + k);
            v2f ax1 = *(const v2f*)(aS1 + kb + k);
            v2f ag0 = *(const v2f*)(aN0 + kb + k);
            v2f ag1 = *(const v2f*)(aN1 + kb + k);
            const int pr = ((k + koff) >> 1) * SP + 2 * (cb + mcol);
#pragma unroll
            for (int ct = 0; ct < 4; ++ct) {
                v2f bs = *(const v2f*)(sWs + pr + 32 * ct);   // half of a ds_load_2addr_b64
                acc[0][ct] = __builtin_amdgcn_wmma_f32_16x16x4_f32(
                    false, ax0, false, bs, (short)0, acc[0][ct], false, false);
                acc[1][ct] = __builtin_amdgcn_wmma_f32_16x16x4_f32(
                    false, ax1, false, bs, (short)0, acc[1][ct], false, false);
                v2f bn = *(const v2f*)(sWn + pr + 32 * ct);
                acc[0][ct] = __builtin_amdgcn_wmma_f32_16x16x4_f32(
                    false, ag0, false, bn, (short)0, acc[0][ct], false, false);
                acc[1][ct] = __builtin_amdgcn_wmma_f32_16x16x4_f32(
                    false, ag1, false, bn, (short)0, acc[1][ct], false, false);
            }
        }
        __syncthreads();
    }

    if (!valid) return;
    // C/D layout: VGPR v = row (tile_row0 + v) for lanes 0-15, (+8+v) for lanes 16-31.
#pragma unroll
    for (int rt = 0; rt < 2; ++rt) {
        const int rbase = row0 + rt * 16 + ((lane >> 4) << 3);
#pragma unroll
        for (int ct = 0; ct < 4; ++ct) {
            int col = cb + ct * 16 + mcol;
            if (MODE == 0) {
                float s = 0.f, sq = 0.f;
#pragma unroll
                for (int v = 0; v < 8; ++v) {
                    float val = acc[rt][ct][v];
                    outp[(size_t)(rbase + v) * D + col] = val;
                    s += val; sq += val * val;
                }
                atomicAdd(&stats[col], s);
                atomicAdd(&stats[D + col], sq);
            } else {
#pragma unroll
                for (int v = 0; v < 8; ++v)
                    outp[(size_t)(rbase + v) * D * NUM_EXPERTS + col * NUM_EXPERTS] =
                        acc[rt][ct][v];
            }
        }
    }
}

// =================== batchnorm ===================

__global__ void bn_finalize(const float* __restrict__ stats, const float* __restrict__ gamma,
                            const float* __restrict__ beta, float* __restrict__ ss) {
    int d = threadIdx.x;  // 128 threads
    float mean = stats[d] * (1.0f / N_NODES);
    float var  = stats[D + d] * (1.0f / N_NODES) - mean * mean;
    float sc   = gamma[d] * rsqrtf(var + 1e-5f);
    ss[d]     = sc;
    ss[D + d] = beta[d] - mean * sc;
}

__global__ void bn_apply(float* __restrict__ h, const float* __restrict__ ss, int n4) {
    int i = blockIdx.x * blockDim.x + threadIdx.x;
    if (i >= n4) return;
    int f = (i << 2) & (D - 1);
    float4 v = ((float4*)h)[i];
    v.x = fmaxf(v.x * ss[f + 0] + ss[D + f + 0], 0.f);
    v.y = fmaxf(v.y * ss[f + 1] + ss[D + f + 1], 0.f);
    v.z = fmaxf(v.z * ss[f + 2] + ss[D + f + 2], 0.f);
    v.w = fmaxf(v.w * ss[f + 3] + ss[D + f + 3], 0.f);
    ((float4*)h)[i] = v;
}

// =================== host orchestration ===================

extern "C" void kernel_launch(void* const* d_in, const int* in_sizes, int n_in,
                              void* d_out, int out_size, void* d_ws, size_t ws_size,
                              hipStream_t stream) {
    const float* x      = (const float*)d_in[0];
    const int*   eidx   = (const int*)d_in[1];      // [2, E]
    const float* W_self = (const float*)d_in[2];    // [8,2,128,128]
    const float* W_nbr  = (const float*)d_in[3];
    const float* bvec   = (const float*)d_in[4];    // [8,2,128]
    const float* gamma  = (const float*)d_in[5];    // [8,1,128]
    const float* beta   = (const float*)d_in[6];
    float* out = (float*)d_out;                     // [N,128,8]

    const int* src = eidx;
    const int* dst = eidx + N_EDGES;

    const size_t ND = (size_t)N_NODES * D;
    float* agg0  = (float*)d_ws;        // N*D  mean-agg of x (shared by all experts)
    float* h     = agg0 + ND;           // N*D  per-expert hidden
    float* agg1  = h + ND;              // N*D  per-expert layer-1 aggregation
    float* stats = agg1 + ND;           // 2*D  colsum / colsumsq
    float* ss    = stats + 2 * D;       // 2*D  scale / shift
    int* cnt     = (int*)(ss + 2 * D);  // N
    int* cursor  = cnt + N_NODES;       // N
    int* rp      = cursor + N_NODES;    // N+1
    int* bsum    = rp + N_NODES + 1;    // NB
    int* boff    = bsum + NB;           // NB
    int* list    = boff + NB;           // E

    const int n4  = (int)(ND / 4);
    const int blk = 256;

    // ---- CSR build (once; reused for all 9 aggregations) ----
    hipMemsetAsync(cnt, 0, N_NODES * sizeof(int), stream);
    hipMemsetAsync(cursor, 0, N_NODES * sizeof(int), stream);
    count_kernel<<<(N_EDGES + blk - 1) / blk, blk, 0, stream>>>(dst, cnt, N_EDGES);
    scan1_kernel<<<NB, blk, 0, stream>>>(cnt, bsum);
    scan2_kernel<<<1, 1, 0, stream>>>(bsum, boff, rp + N_NODES);
    scan3_kernel<<<NB, blk, 0, stream>>>(cnt, boff, rp);
    fill_kernel<<<(N_EDGES + blk - 1) / blk, blk, 0, stream>>>(src, dst, rp, cursor, list,
                                                               N_EDGES);

    // ---- shared layer-0 mean aggregation of x ----
    gather_mean<<<(N_NODES * 32) / blk, blk, 0, stream>>>(x, list, rp, agg0);

    const int gemm_grid = (N_TASKS + 3) / 4;  // 782 blocks: 4 double-row tasks x 2 col halves
    for (int e = 0; e < NUM_EXPERTS; ++e) {
        const float* Ws0 = W_self + ((size_t)e * 2 + 0) * D * D;
        const float* Wn0 = W_nbr  + ((size_t)e * 2 + 0) * D * D;
        const float* Ws1 = W_self + ((size_t)e * 2 + 1) * D * D;
        const float* Wn1 = W_nbr  + ((size_t)e * 2 + 1) * D * D;
        const float* b0  = bvec + ((size_t)e * 2 + 0) * D;
        const float* b1  = bvec + ((size_t)e * 2 + 1) * D;

        // layer 0: h = x@Ws0 + agg0@Wn0 + b0, fused BN-stat accumulation
        hipMemsetAsync(stats, 0, 2 * D * sizeof(float), stream);
        sage_gemm<0><<<gemm_grid, blk, 0, stream>>>(x, agg0, Ws0, Wn0, b0, h, stats);
        bn_finalize<<<1, D, 0, stream>>>(stats, gamma + (size_t)e * D, beta + (size_t)e * D, ss);
        bn_apply<<<(n4 + blk - 1) / blk, blk, 0, stream>>>(h, ss, n4);

        // layer-1 aggregation of h (gather, no atomics, inv-deg fused)
        gather_mean<<<(N_NODES * 32) / blk, blk, 0, stream>>>(h, list, rp, agg1);

        // layer 1: out[:,:,e] = h@Ws1 + agg1@Wn1 + b1  (strided store)
        sage_gemm<1><<<gemm_grid, blk, 0, stream>>>(h, agg1, Ws1, Wn1, b1, out + e, nullptr);
    }
    (void)in_sizes; (void)n_in; (void)out_size; (void)ws_size;
}